// BioRNN_19035295056367
// MI455X (gfx1250) — compile-verified
//
#include <hip/hip_runtime.h>
#include <hip/hip_bf16.h>

// Problem dims (compile-time constants from the reference)
#define B_    64
#define T_    2048
#define I_    128
#define H_    512
#define O_    128
#define KTOT  640            // I + H, fused [x_t | h] K dimension
#define NWG   16             // persistent workgroups (one per WGP, always resident)
#define TPB   256            // 8 wave32 per WG
#define WREC_PITCH 644       // 644 % 64 == 4  -> conflict-free column reads
#define WFC_PITCH  516       // 516 % 64 == 4

typedef float v2f __attribute__((ext_vector_type(2)));
typedef float v8f __attribute__((ext_vector_type(8)));

// ---------------------------------------------------------------------------
// Prep: masked+concatenated weights, summed bias, h0 copy (into buffer 2 of
// the triple-buffered h ring), barrier reset. Runs every call -> replay safe.
// ---------------------------------------------------------------------------
__global__ void biornn_prep(const float* __restrict__ Wih, const float* __restrict__ bih,
                            const float* __restrict__ Whh, const float* __restrict__ bhh,
                            const float* __restrict__ mih, const float* __restrict__ mhh,
                            const float* __restrict__ h0,
                            float* __restrict__ Wcat, float* __restrict__ bias,
                            float* __restrict__ h_bufs, unsigned* __restrict__ sync) {
    int idx = blockIdx.x * blockDim.x + threadIdx.x;
    if (idx < H_ * KTOT) {
        int h = idx / KTOT, k = idx - h * KTOT;
        float v;
        if (k < I_) v = Wih[h * I_ + k] * mih[h * I_ + k];
        else        v = Whh[h * H_ + (k - I_)] * mhh[h * H_ + (k - I_)];
        Wcat[idx] = v;
    }
    if (idx < H_)       bias[idx] = bih[idx] + bhh[idx];
    if (idx < B_ * H_)  h_bufs[2 * B_ * H_ + idx] = h0[idx];   // ring slot 2 = h_{-1}
    if (idx < 8)        sync[idx] = 0u;
}

// ---------------------------------------------------------------------------
// Persistent recurrence kernel: 2048 steps, one *split-phase* device barrier
// per step. Readout of out[:,t-1,:] + x prefetch execute in the signal->wait
// window (legal: h_{t-1} was made visible by the previous barrier), hiding
// them under the barrier latency. Triple-buffered h gives the 3-step reuse
// distance that makes the overlapped readout race-free.
// ---------------------------------------------------------------------------
__launch_bounds__(TPB, 1)
__global__ void biornn_persistent(const float* __restrict__ x,
                                  const float* __restrict__ Wcat,
                                  const float* __restrict__ bias,
                                  const float* __restrict__ Wfc,
                                  const float* __restrict__ bfc,
                                  float* __restrict__ h_bufs,
                                  float* __restrict__ out,
                                  unsigned* __restrict__ sync) {
    extern __shared__ float lds[];
    float* wrec = lds;                        // [32][WREC_PITCH] masked [Wih|Whh] slice
    float* wfcl = lds + 32 * WREC_PITCH;      // [16][WFC_PITCH]  W_fc slice

    const int g    = blockIdx.x;
    const int tid  = threadIdx.x;
    const int lane = tid & 31;
    const int wid  = tid >> 5;
    const int half = lane >> 4;               // 0: lanes 0-15, 1: lanes 16-31
    const int lr   = lane & 15;

    const int n0g = g * 32;                   // this WG's h-column base
    const int og  = (g & 7) * 16;             // readout O-column base
    const int mg  = (g >> 3) * 32;            // readout batch-row base

    // Stage weight slices into LDS (padded pitch -> bank-conflict-free B reads)
    for (int i = tid; i < 32 * KTOT; i += TPB) {
        int r = i / KTOT, k = i - r * KTOT;
        wrec[r * WREC_PITCH + k] = Wcat[(size_t)(n0g + r) * KTOT + k];
    }
    for (int i = tid; i < 16 * H_; i += TPB) {
        int r = i >> 9, k = i & (H_ - 1);
        wfcl[r * WFC_PITCH + k] = Wfc[(size_t)(og + r) * H_ + k];
    }
    __syncthreads();

    // Per-wave recurrence tile: 4 row-tiles x 2 col-tiles
    const int m0    = (wid & 3) * 16;                 // batch-row tile base
    const int nl    = (wid >> 2) * 16;                // local h-col tile base
    const int nglob = n0g + nl;
    const float  bv      = bias[nglob + lr];          // column bias (hoisted)
    const float* wr_base = &wrec[(nl + lr) * WREC_PITCH];

    // Readout setup (waves 0,1 each take one 16x16 out tile)
    const float  bfc_v   = bfc[og + lr];
    const float* wf_base = &wfcl[lr * WFC_PITCH];
    const int    mread   = mg + (wid & 1) * 16;

    // out[:, tt, :] = h_tt @ Wfc^T + b_fc  (waves 0,1 only; wave-uniform branch)
    auto do_readout = [&](const float* __restrict__ hbase, int tt) {
        v8f racc = {};
        const float* hr = hbase + (size_t)(mread + lr) * H_;
        #pragma unroll 4
        for (int k0 = 0; k0 < H_; k0 += 4) {
            v2f a = *(const v2f*)(hr      + k0 + 2 * half);
            v2f b = *(const v2f*)(wf_base + k0 + 2 * half);
            racc = __builtin_amdgcn_wmma_f32_16x16x4_f32(false, a, false, b,
                                                         (short)0, racc, false, false);
        }
        #pragma unroll
        for (int v = 0; v < 8; ++v) {
            int brow = mread + 8 * half + v;
            out[((size_t)brow * T_ + tt) * O_ + og + lr] = racc[v] + bfc_v;
        }
    };

    // Triple-buffered h ring: step t writes bufA, reads bufC (=h_{t-1});
    // rotation (A,B,C) -> (B,C,A) each step.
    float* bufA = h_bufs;                     // written at t = 0,3,6,...
    float* bufB = h_bufs + B_ * H_;           // written at t = 1,4,7,...
    float* bufC = h_bufs + 2 * B_ * H_;       // holds h0; written at t = 2,5,...

    unsigned* cnt = &sync[0];
    unsigned* gen = &sync[1];
    unsigned local_gen = 0;

    const float* xrow_base = x + (size_t)(m0 + lr) * T_ * I_;  // A-row for this lane

    for (int t = 0; t < T_; ++t) {
        float*       hwrite = bufA;
        const float* hread  = bufC;           // h_{t-1}
        v8f acc = {};

        // ---- K = 0..127 : input projection, A from x (global/L2) ----
        const float* xrow = xrow_base + (size_t)t * I_;
        #pragma unroll 4
        for (int k0 = 0; k0 < I_; k0 += 4) {
            v2f a = *(const v2f*)(xrow    + k0 + 2 * half);
            v2f b = *(const v2f*)(wr_base + k0 + 2 * half);
            acc = __builtin_amdgcn_wmma_f32_16x16x4_f32(false, a, false, b,
                                                        (short)0, acc, false, false);
        }
        // ---- K = 128..639 : recurrence, A from h_{t-1} (global/L2) ----
        const float* hrow    = hread + (m0 + lr) * H_;
        const float* wh_base = wr_base + I_;
        #pragma unroll 4
        for (int k0 = 0; k0 < H_; k0 += 4) {
            v2f a = *(const v2f*)(hrow    + k0 + 2 * half);
            v2f b = *(const v2f*)(wh_base + k0 + 2 * half);
            acc = __builtin_amdgcn_wmma_f32_16x16x4_f32(false, a, false, b,
                                                        (short)0, acc, false, false);
        }

        // tanh + scatter per C-matrix layout: lane<16 -> M=m0+v, lane>=16 -> M=m0+8+v
        {
            float* hcol = hwrite + (size_t)(m0 + 8 * half) * H_ + nglob + lr;
            #pragma unroll
            for (int v = 0; v < 8; ++v)
                hcol[(size_t)v * H_] = tanhf(acc[v] + bv);
        }

        // ---- split-phase device barrier: SIGNAL ----
        __syncthreads();                       // all waves' h stores issued
        if (tid == 0) {
            __threadfence();                   // release h_t stores
            unsigned prev = atomicAdd(cnt, 1u);
            if (prev == NWG - 1u) {
                atomicExch(cnt, 0u);
                __threadfence();
                atomicAdd(gen, 1u);            // release the barrier
            }
        }

        // ---- post-signal window: independent work, hidden under sync ----
        if (t > 0 && wid < 2)
            do_readout(hread, t - 1);          // h_{t-1}: visible since barrier t-1
        if (t + 1 < T_) {                      // prefetch next step's x rows
            const float* xn = xrow_base + (size_t)(t + 1) * I_;
            #pragma unroll
            for (int j = 0; j < 4; ++j)
                __builtin_prefetch(xn + j * 32, 0, 0);
        }

        // ---- WAIT ----
        if (tid == 0) {
            while (__atomic_load_n(gen, __ATOMIC_RELAXED) == local_gen)
                __builtin_amdgcn_s_sleep(1);
            __threadfence();                   // acquire other WGs' h_t stores
        }
        local_gen++;
        __syncthreads();

        // rotate ring (A,B,C) -> (B,C,A)
        float* tmp = bufA; bufA = bufB; bufB = bufC; bufC = tmp;
    }

    // Final readout: h_{T-1} is in bufC after the last rotation.
    if (wid < 2)
        do_readout(bufC, T_ - 1);
}

// ---------------------------------------------------------------------------
extern "C" void kernel_launch(void* const* d_in, const int* in_sizes, int n_in,
                              void* d_out, int out_size, void* d_ws, size_t ws_size,
                              hipStream_t stream) {
    (void)in_sizes; (void)n_in; (void)out_size; (void)ws_size;

    const float* x    = (const float*)d_in[0];
    const float* h0   = (const float*)d_in[1];
    const float* Wih  = (const float*)d_in[2];
    const float* bih  = (const float*)d_in[3];
    const float* Whh  = (const float*)d_in[4];
    const float* bhh  = (const float*)d_in[5];
    const float* mih  = (const float*)d_in[6];
    const float* mhh  = (const float*)d_in[7];
    const float* Wfc  = (const float*)d_in[8];
    const float* bfc  = (const float*)d_in[9];
    float*       out  = (float*)d_out;

    // Workspace layout (floats): Wcat | bias | h_ring[3] | sync
    float* ws     = (float*)d_ws;
    float* Wcat   = ws;                        // 512*640
    float* bias   = Wcat + H_ * KTOT;          // 512
    float* h_bufs = bias + H_;                 // 3 * 64*512
    unsigned* sync = (unsigned*)(h_bufs + 3 * B_ * H_);

    const int prep_n      = H_ * KTOT;                        // 327680
    const int prep_blocks = (prep_n + TPB - 1) / TPB;
    biornn_prep<<<prep_blocks, TPB, 0, stream>>>(Wih, bih, Whh, bhh, mih, mhh, h0,
                                                 Wcat, bias, h_bufs, sync);

    const size_t lds_bytes = (size_t)(32 * WREC_PITCH + 16 * WFC_PITCH) * sizeof(float);
    biornn_persistent<<<NWG, TPB, lds_bytes, stream>>>(x, Wcat, bias, Wfc, bfc,
                                                       h_bufs, out, sync);
}